// GVPStructureEncoder_31147102831084
// MI455X (gfx1250) — compile-verified
//
#include <hip/hip_runtime.h>

typedef __attribute__((ext_vector_type(2))) float v2f;
typedef __attribute__((ext_vector_type(8))) float v8f;

#define EPSF 1e-8f

__device__ __forceinline__ v8f wmma4(v2f a, v2f b, v8f c) {
  // V_WMMA_F32_16X16X4_F32: D = A(16x4) * B(4x16) + C, fp32 exact
  return __builtin_amdgcn_wmma_f32_16x16x4_f32(
      /*neg_a=*/false, a, /*neg_b=*/false, b,
      /*c_mod=*/(short)0, c, /*reuse_a=*/false, /*reuse_b=*/false);
}

// One wave32 computes D(16 x wN) = A(16 x K) @ W(K x wN) [+bias][relu].
// A: row-major (LDS), leading dim lda (even). W: row-major global (K x wN).
// out: row-major, leading dim ldo. K must be a multiple of 4.
template <bool HAS_BIAS, bool RELU>
__device__ __forceinline__ void gemm16(const float* A, int lda,
                                       const float* __restrict__ W, int wN, int K,
                                       float* out, int ldo,
                                       const float* __restrict__ bias) {
  const int lane = threadIdx.x;        // blockDim.x == 32
  const int row  = lane & 15;          // M index for A frag, N index for B frag
  const int kk   = (lane >> 4) << 1;   // lanes 0-15 -> K pair {0,1}; 16-31 -> {2,3}
  const int half = lane >> 4;
  for (int n0 = 0; n0 < wN; n0 += 16) {
    v8f acc = {};
    for (int k0 = 0; k0 < K; k0 += 4) {
      // A fragment: adjacent pair, 8B aligned -> single ds_load_b64
      v2f af = *(const v2f*)&A[row * lda + k0 + kk];
      v2f bf;
      bf.x = W[(k0 + kk) * wN + n0 + row];
      bf.y = W[(k0 + kk + 1) * wN + n0 + row];
      acc = wmma4(af, bf, acc);
    }
    float bv = HAS_BIAS ? bias[n0 + row] : 0.0f;   // invariant across r
    // C/D layout: VGPR r -> rows r (lanes 0-15) and r+8 (lanes 16-31), col = lane&15
    for (int r = 0; r < 8; ++r) {
      float v = acc[r];
      if (HAS_BIAS) v += bv;
      if (RELU) v = fmaxf(v, 0.0f);
      out[(r + 8 * half) * ldo + n0 + row] = v;
    }
  }
}

// ---------------- input projection: h_s = relu(seq @ Win + b) ------------
__global__ void gvp_input_proj(const float* __restrict__ seq,
                               const float* __restrict__ W,
                               const float* __restrict__ b,
                               float* __restrict__ hs) {
  __shared__ float a[16 * 20];
  const int t0 = blockIdx.x * 16;
  const int lane = threadIdx.x;
  for (int idx = lane; idx < 16 * 20; idx += 32) {
    int e = idx / 20, c = idx % 20;
    a[idx] = seq[(t0 + e) * 20 + c];
  }
  __syncthreads();
  gemm16<true, true>(a, 20, W, 128, 20, hs + (size_t)t0 * 128, 128, b);
}

// ---------------- helpers ------------------------------------------------
__global__ void gvp_zero_f(float* p, long n) {
  long i = (long)blockIdx.x * blockDim.x + threadIdx.x;
  if (i < n) p[i] = 0.0f;
}
__global__ void gvp_zero_i(int* p, int n) {
  int i = blockIdx.x * blockDim.x + threadIdx.x;
  if (i < n) p[i] = 0;
}
__global__ void gvp_count_deg(const int* __restrict__ ei, int E, int* __restrict__ deg) {
  int e = blockIdx.x * blockDim.x + threadIdx.x;
  if (e < E) atomicAdd(&deg[ei[E + e]], 1);
}
__global__ void gvp_finalize(float* __restrict__ s, float* __restrict__ v,
                             const int* __restrict__ deg, int N) {
  long i = (long)blockIdx.x * blockDim.x + threadIdx.x;
  long ns = (long)N * 128;
  if (i < ns) {
    float d = fmaxf((float)deg[i >> 7], 1.0f);
    s[i] /= d;
  } else {
    long k = i - ns;
    if (k < (long)N * 48) {
      float d = fmaxf((float)deg[k / 48], 1.0f);
      v[k] /= d;
    }
  }
}

// ---------------- GVP with si=128, vi=vo=16 (g1 / g2) --------------------
template <bool ACT>
__device__ __forceinline__ void gvp128(float* ms /*16x144*/, float* sm /*16x128 in/out*/,
                                       float (*vo)[16 * 16] /*3x, in/out*/,
                                       float (*vh)[16 * 32] /*3x scratch, ld 32*/,
                                       const float* wh, const float* wv,
                                       const float* wsw, const float* wsb) {
  const int lane = threadIdx.x;
  // vh = vo @ wh (16x16), per coordinate
  for (int c = 0; c < 3; ++c)
    gemm16<false, false>(vo[c], 16, wh, 16, 16, vh[c], 32, nullptr);
  __syncthreads();
  // ms = [sm | ||vh||]
  for (int idx = lane; idx < 16 * 128; idx += 32) {
    int e = idx >> 7, col = idx & 127;
    ms[e * 144 + col] = sm[idx];
  }
  for (int idx = lane; idx < 16 * 16; idx += 32) {
    int e = idx >> 4, h = idx & 15;
    float s2 = 0.0f;
    for (int c = 0; c < 3; ++c) { float x = vh[c][e * 32 + h]; s2 += x * x; }
    ms[e * 144 + 128 + h] = sqrtf(fmaxf(s2, EPSF));
  }
  __syncthreads();
  gemm16<true, ACT>(ms, 144, wsw, 128, 144, sm, 128, wsb);
  for (int c = 0; c < 3; ++c)
    gemm16<false, false>(vh[c], 32, wv, 16, 16, vo[c], 16, nullptr);
  __syncthreads();
  if (ACT) {  // vector gate: v *= sigmoid(||v||)
    for (int idx = lane; idx < 16 * 16; idx += 32) {
      float s2 = 0.0f;
      for (int c = 0; c < 3; ++c) { float x = vo[c][idx]; s2 += x * x; }
      float g = 1.0f / (1.0f + __expf(-sqrtf(fmaxf(s2, EPSF))));
      for (int c = 0; c < 3; ++c) vo[c][idx] *= g;
    }
    __syncthreads();
  }
}

// ---------------- edge message + aggregation kernel ----------------------
__global__ void gvp_edge_conv(const float* __restrict__ hs, const float* __restrict__ hv,
                              const int* __restrict__ ei, int E,
                              const float* wh0, const float* wv0, const float* wsw0, const float* wsb0,
                              const float* wh1, const float* wv1, const float* wsw1, const float* wsb1,
                              const float* wh2, const float* wv2, const float* wsw2, const float* wsb2,
                              float* __restrict__ aggs, float* __restrict__ aggv) {
  __shared__ float ms[16 * 288];      // [s_j | s_i | vn]
  __shared__ float mv[3][16 * 32];    // per-coord vector channels (vi=32)
  __shared__ float vh[3][16 * 32];    // vh scratch (ld 32)
  __shared__ float sm[16 * 128];      // scalar message
  __shared__ float vo[3][16 * 16];    // vector message

  const int lane = threadIdx.x;
  const int te = blockIdx.x * 16;

  // ---- gather (pad tail edges with node 0; they are never aggregated) ----
  for (int e = 0; e < 16; ++e) {
    int ge = te + e;
    int j = 0, i = 0;
    if (ge < E) { j = ei[ge]; i = ei[E + ge]; }
    for (int t = 0; t < 4; ++t) {
      int c = lane + 32 * t;
      ms[e * 288 + c]       = hs[(size_t)j * 128 + c];
      ms[e * 288 + 128 + c] = hs[(size_t)i * 128 + c];
    }
    int src = (lane < 16) ? j : i;
    int h = lane & 15;
    for (int c = 0; c < 3; ++c)
      mv[c][e * 32 + lane] = hv[((size_t)src * 16 + h) * 3 + c];
  }
  __syncthreads();

  // ---- g0: si=256(+32 vn), vi=32 -> so=128, vo=16 ----
  for (int c = 0; c < 3; ++c)
    gemm16<false, false>(mv[c], 32, wh0, 32, 32, vh[c], 32, nullptr);
  __syncthreads();
  for (int idx = lane; idx < 16 * 32; idx += 32) {
    int e = idx >> 5, h = idx & 31;
    float s2 = 0.0f;
    for (int c = 0; c < 3; ++c) { float x = vh[c][e * 32 + h]; s2 += x * x; }
    ms[e * 288 + 256 + h] = sqrtf(fmaxf(s2, EPSF));
  }
  __syncthreads();
  gemm16<true, true>(ms, 288, wsw0, 128, 288, sm, 128, wsb0);
  for (int c = 0; c < 3; ++c)
    gemm16<false, false>(vh[c], 32, wv0, 16, 32, vo[c], 16, nullptr);
  __syncthreads();
  for (int idx = lane; idx < 16 * 16; idx += 32) {
    float s2 = 0.0f;
    for (int c = 0; c < 3; ++c) { float x = vo[c][idx]; s2 += x * x; }
    float g = 1.0f / (1.0f + __expf(-sqrtf(fmaxf(s2, EPSF))));
    for (int c = 0; c < 3; ++c) vo[c][idx] *= g;
  }
  __syncthreads();

  // ---- g1 (relu+gate), g2 (no activations) ----
  gvp128<true >(ms, sm, vo, vh, wh1, wv1, wsw1, wsb1);
  gvp128<false>(ms, sm, vo, vh, wh2, wv2, wsw2, wsb2);

  // ---- mean aggregation (sum via atomics; finalize divides by deg) ----
  for (int e = 0; e < 16; ++e) {
    int ge = te + e;
    if (ge >= E) continue;          // wave-uniform
    int i = ei[E + ge];
    for (int t = 0; t < 4; ++t) {
      int c = lane + 32 * t;
      atomicAdd(&aggs[(size_t)i * 128 + c], sm[e * 128 + c]);
    }
    for (int idx = lane; idx < 48; idx += 32) {
      int h = idx / 3, c = idx % 3;
      atomicAdd(&aggv[((size_t)i * 16 + h) * 3 + c], vo[c][e * 16 + h]);
    }
  }
}

// ---------------- host-side orchestration --------------------------------
extern "C" void kernel_launch(void* const* d_in, const int* in_sizes, int n_in,
                              void* d_out, int out_size, void* d_ws, size_t ws_size,
                              hipStream_t stream) {
  const int N = 8 * 2048;
  const float* seq  = (const float*)d_in[0];
  const float* in_w = (const float*)d_in[2];   // (20,128)
  const float* in_b = (const float*)d_in[3];   // (128,)
  const int*   ei   = (const int*)d_in[n_in - 1];
  const int    E    = in_sizes[n_in - 1] / 2;

  float* hsA = (float*)d_ws;
  float* hvA = hsA + (size_t)N * 128;
  float* hsB = hvA + (size_t)N * 48;
  float* hvB = hsB + (size_t)N * 128;
  int*   deg = (int*)(hvB + (size_t)N * 48);

  // degrees (computed once; graph is static)
  gvp_zero_i<<<(N + 255) / 256, 256, 0, stream>>>(deg, N);
  gvp_count_deg<<<(E + 255) / 256, 256, 0, stream>>>(ei, E, deg);

  // input GVP: scalar linear + relu; vectors start at zero
  gvp_input_proj<<<N / 16, 32, 0, stream>>>(seq, in_w, in_b, hsA);
  gvp_zero_f<<<(int)(((long)N * 48 + 255) / 256), 256, 0, stream>>>(hvA, (long)N * 48);

  float* curS = hsA;
  float* curV = hvA;
  for (int l = 0; l < 3; ++l) {
    const float* p[12];
    for (int k = 0; k < 12; ++k) p[k] = (const float*)d_in[4 + l * 12 + k];
    float* outS = (l == 2) ? (float*)d_out : ((curS == hsA) ? hsB : hsA);
    float* outV = (curV == hvA) ? hvB : hvA;

    gvp_zero_f<<<(int)(((long)N * 128 + 255) / 256), 256, 0, stream>>>(outS, (long)N * 128);
    gvp_zero_f<<<(int)(((long)N * 48 + 255) / 256), 256, 0, stream>>>(outV, (long)N * 48);

    gvp_edge_conv<<<(E + 15) / 16, 32, 0, stream>>>(curS, curV, ei, E,
        p[0], p[1], p[2], p[3],
        p[4], p[5], p[6], p[7],
        p[8], p[9], p[10], p[11],
        outS, outV);

    gvp_finalize<<<(int)(((long)N * (128 + 48) + 255) / 256), 256, 0, stream>>>(outS, outV, deg, N);
    curS = outS;
    curV = outV;
  }
}